// CFConv_38826504356688
// MI455X (gfx1250) — compile-verified
//
#include <hip/hip_runtime.h>

#define N_ATOMS 50000
#define N_EDGES 1600000
#define DFEAT   128

typedef float v2f __attribute__((ext_vector_type(2)));
typedef float v4f __attribute__((ext_vector_type(4)));
typedef float v8f __attribute__((ext_vector_type(8)));

// ---------------------------------------------------------------------------
// Kernel 1: CSR row offsets from sorted seg_i via lower_bound (deterministic,
// lets the edge kernel run atomic-free).
// ---------------------------------------------------------------------------
__global__ void cfconv_offsets_kernel(const int* __restrict__ seg,
                                      int* __restrict__ offs) {
    int a = blockIdx.x * blockDim.x + threadIdx.x;
    if (a > N_ATOMS) return;
    int lo = 0, hi = N_EDGES;
    while (lo < hi) {
        int mid = (lo + hi) >> 1;
        if (seg[mid] < a) lo = mid + 1; else hi = mid;
    }
    offs[a] = lo;
}

// ---------------------------------------------------------------------------
// Kernel 2: f = x @ W1  (fp32 GEMM via V_WMMA_F32_16X16X4_F32)
// One wave computes one 16x16 tile; K-loop of 32 WMMAs.
// A layout (16x4 f32): lane L holds M=(L&15); v0: K=k+2*(L>>4), v1: K+1.
// B layout (4x16 f32): lane L holds N=n0+(L&15); v0: K=k+2*(L>>4), v1: K+1.
// C/D layout: vgpr r -> M = r + 8*(L>>4), N = (L&15).
// ---------------------------------------------------------------------------
__global__ void cfconv_gemm1_kernel(const float* __restrict__ X,
                                    const float* __restrict__ W,
                                    float* __restrict__ F) {
    const int gid  = blockIdx.x * 8 + (threadIdx.x >> 5);   // global wave id
    const int lane = threadIdx.x & 31;
    const int tm = gid >> 3;          // 0..3124  (50000/16 row tiles)
    const int tn = gid & 7;           // 0..7     (128/16 col tiles)
    const int m0 = tm * 16, n0 = tn * 16;
    const int hl = lane >> 4;         // half-wave select
    const int l  = lane & 15;

    const float* xrow = X + (size_t)(m0 + l) * DFEAT + 2 * hl;
    const float* wcol = W + (size_t)(2 * hl) * DFEAT + n0 + l;

    v8f c = {};
    #pragma unroll
    for (int k = 0; k < DFEAT; k += 4) {
        v2f a = *(const v2f*)(xrow + k);                 // contiguous K pair
        v2f b;
        b.x = wcol[(size_t)k * DFEAT];
        b.y = wcol[(size_t)(k + 1) * DFEAT];
        c = __builtin_amdgcn_wmma_f32_16x16x4_f32(false, a, false, b,
                                                  (short)0, c, false, false);
    }
    float* out = F + (size_t)(m0 + 8 * hl) * DFEAT + n0 + l;
    #pragma unroll
    for (int r = 0; r < 8; ++r) out[(size_t)r * DFEAT] = c[r];
}

// ---------------------------------------------------------------------------
// Kernel 3: conv[i,:] = sum_{e in rows(i)} w_ij[e,:] * f[idx_j[e],:]
// One wave per atom: 32 lanes x float4 = 128-wide row. The 819 MB w_ij
// stream (touched once) uses non-temporal loads so it does not evict the
// L2-resident f gather table; idx_j likewise. 2-edge unroll with dual
// accumulators doubles loads in flight. No atomics -> deterministic.
// ---------------------------------------------------------------------------
__global__ void cfconv_edge_kernel(const float* __restrict__ w_ij,
                                   const float* __restrict__ f,
                                   const int* __restrict__ idx_j,
                                   const int* __restrict__ offs,
                                   float* __restrict__ conv) {
    const int wave = threadIdx.x >> 5;
    const int lane = threadIdx.x & 31;
    const int atom = blockIdx.x * 8 + wave;   // grid sized exactly: no divergence
    const int e0 = offs[atom];
    const int e1 = offs[atom + 1];
    const size_t coff = (size_t)lane * 4;

    v4f acc0 = {0.f, 0.f, 0.f, 0.f};
    v4f acc1 = {0.f, 0.f, 0.f, 0.f};

    int e = e0;
    for (; e + 2 <= e1; e += 2) {
        const int j0 = __builtin_nontemporal_load(idx_j + e);
        const int j1 = __builtin_nontemporal_load(idx_j + e + 1);
        const v4f w0 = __builtin_nontemporal_load(
            (const v4f*)(w_ij + (size_t)e * DFEAT + coff));
        const v4f w1 = __builtin_nontemporal_load(
            (const v4f*)(w_ij + (size_t)(e + 1) * DFEAT + coff));
        const v4f f0 = *(const v4f*)(f + (size_t)j0 * DFEAT + coff);
        const v4f f1 = *(const v4f*)(f + (size_t)j1 * DFEAT + coff);
        acc0 = __builtin_elementwise_fma(w0, f0, acc0);
        acc1 = __builtin_elementwise_fma(w1, f1, acc1);
    }
    if (e < e1) {
        const int j0 = __builtin_nontemporal_load(idx_j + e);
        const v4f w0 = __builtin_nontemporal_load(
            (const v4f*)(w_ij + (size_t)e * DFEAT + coff));
        const v4f f0 = *(const v4f*)(f + (size_t)j0 * DFEAT + coff);
        acc0 = __builtin_elementwise_fma(w0, f0, acc0);
    }
    acc0 += acc1;
    *(v4f*)(conv + (size_t)atom * DFEAT + coff) = acc0;
}

// ---------------------------------------------------------------------------
// Kernel 4: OUT = OUT @ W2 + b2, in place.
// Each block stages its 16x128 input strip in LDS (removes the in-place RAW
// hazard across n-tiles), then 8 waves each compute one 16x16 WMMA tile.
// ---------------------------------------------------------------------------
__global__ void cfconv_gemm2_kernel(float* __restrict__ OUT,
                                    const float* __restrict__ W2,
                                    const float* __restrict__ b2) {
    __shared__ float As[16 * DFEAT];
    const int m0 = blockIdx.x * 16;

    // cooperative strip load: 2048 floats / 256 threads = 2 float4 each
    {
        const v4f* src = (const v4f*)(OUT + (size_t)m0 * DFEAT);
        v4f* dst = (v4f*)As;
        #pragma unroll
        for (int i = threadIdx.x; i < 16 * DFEAT / 4; i += 256) dst[i] = src[i];
    }
    __syncthreads();

    const int wave = threadIdx.x >> 5;
    const int lane = threadIdx.x & 31;
    const int n0 = wave * 16;
    const int hl = lane >> 4;
    const int l  = lane & 15;

    const float* arow = As + l * DFEAT + 2 * hl;
    const float* wcol = W2 + (size_t)(2 * hl) * DFEAT + n0 + l;

    v8f c = {};
    #pragma unroll
    for (int k = 0; k < DFEAT; k += 4) {
        v2f a = *(const v2f*)(arow + k);
        v2f b;
        b.x = wcol[(size_t)k * DFEAT];
        b.y = wcol[(size_t)(k + 1) * DFEAT];
        c = __builtin_amdgcn_wmma_f32_16x16x4_f32(false, a, false, b,
                                                  (short)0, c, false, false);
    }
    const float bias = b2[n0 + l];
    float* out = OUT + (size_t)(m0 + 8 * hl) * DFEAT + n0 + l;
    #pragma unroll
    for (int r = 0; r < 8; ++r) out[(size_t)r * DFEAT] = c[r] + bias;
}

// ---------------------------------------------------------------------------
extern "C" void kernel_launch(void* const* d_in, const int* in_sizes, int n_in,
                              void* d_out, int out_size, void* d_ws, size_t ws_size,
                              hipStream_t stream) {
    const float* x    = (const float*)d_in[0];
    const float* w_ij = (const float*)d_in[1];
    const int*   seg  = (const int*)d_in[2];
    const int*   idxj = (const int*)d_in[3];
    // d_in[4] = seg_i_sum scalar (known constant N_ATOMS), unused
    const float* W1   = (const float*)d_in[5];
    const float* W2   = (const float*)d_in[6];
    const float* b2   = (const float*)d_in[7];
    float* out = (float*)d_out;

    float* f   = (float*)d_ws;                                    // 25.6 MB
    int*  offs = (int*)((char*)d_ws +
                        (size_t)N_ATOMS * DFEAT * sizeof(float)); // +200 KB

    cfconv_offsets_kernel<<<(N_ATOMS + 1 + 255) / 256, 256, 0, stream>>>(seg, offs);
    cfconv_gemm1_kernel<<<N_ATOMS / 16, 256, 0, stream>>>(x, W1, f);      // 3125 blocks
    cfconv_edge_kernel<<<N_ATOMS / 8, 256, 0, stream>>>(w_ij, f, idxj, offs, out);
    cfconv_gemm2_kernel<<<N_ATOMS / 16, 256, 0, stream>>>(out, W2, b2);
}